// Memory_10041633538600
// MI455X (gfx1250) — compile-verified
//
#include <hip/hip_runtime.h>
#include <hip/hip_bf16.h>

// Shapes
#define Bn   16
#define Dd   512
#define HWn  1024            // H*W
#define Nn   16384           // B*H*W
#define Mm   1024

typedef float v2f __attribute__((ext_vector_type(2)));
typedef float v8f __attribute__((ext_vector_type(8)));

// ---- sortable-uint encoding for exact float atomic max (order-independent, deterministic)
__device__ __forceinline__ unsigned enc_f(float v) {
    unsigned u = __float_as_uint(v);
    return (u & 0x80000000u) ? ~u : (u | 0x80000000u);
}
__device__ __forceinline__ float dec_f(unsigned u) {
    return __uint_as_float((u & 0x80000000u) ? (u ^ 0x80000000u) : ~u);
}

// ---------------------------------------------------------------- K0: init colmax
__global__ void k_init(unsigned* __restrict__ colmax_u) {
    colmax_u[blockIdx.x * 256 + threadIdx.x] = 0u;   // below enc of any float
}

// ---------------------------------------------------------------- K1: normalize + NCHW->(n,d)
// block: 256 thr handles 32 hw positions of one batch, grid = N/32 = 512
__global__ __launch_bounds__(256) void k_normalize(const float* __restrict__ query,
                                                   float* __restrict__ q) {
    __shared__ float part[256];
    __shared__ float inv[32];
    __shared__ __align__(16) float tile[32 * 129];
    const int t = threadIdx.x;
    const int b = blockIdx.x >> 5;
    const int hwb = (blockIdx.x & 31) * 32;
    const int hw_l = t & 31, dt = t >> 5;           // load map: 32 hw lanes x 8 d-groups
    const float* src = query + (size_t)b * (Dd * HWn);

    float acc = 0.f;
    #pragma unroll 8
    for (int it = 0; it < 64; ++it) {
        int d = dt + it * 8;
        float v = src[(size_t)d * HWn + hwb + hw_l];
        acc += v * v;
    }
    part[t] = acc;
    __syncthreads();
    if (t < 32) {
        float s = 0.f;
        #pragma unroll
        for (int k = 0; k < 8; ++k) s += part[k * 32 + t];
        inv[t] = 1.0f / fmaxf(sqrtf(s), 1e-12f);
    }
    __syncthreads();

    const int d_l = t & 31, hw_s = t >> 5;          // store map: 32 d lanes x 8 hw-groups
    const int nb = blockIdx.x * 32;
    for (int dc = 0; dc < Dd; dc += 128) {
        #pragma unroll
        for (int it = 0; it < 16; ++it) {
            int d = dt + it * 8;
            tile[hw_l * 129 + d] = src[(size_t)(dc + d) * HWn + hwb + hw_l];
        }
        __syncthreads();
        #pragma unroll
        for (int ih = 0; ih < 4; ++ih) {
            int hw = hw_s + ih * 8;
            float sc = inv[hw];
            size_t rb = (size_t)(nb + hw) * Dd + dc;
            #pragma unroll
            for (int id = 0; id < 4; ++id) {
                int d = d_l + id * 32;
                q[rb + d] = tile[hw * 129 + d] * sc;
            }
        }
        __syncthreads();
    }
}

// ---------------------------------------------------------------- K2: score = q @ keys^T  (fp32 WMMA)
// block tile 64n x 128m, 8 waves as 2(n) x 4(m), each wave 2x2 frags of 16x16, BK=32
__global__ __launch_bounds__(256) void k_gemm_score(const float* __restrict__ q,
                                                    const float* __restrict__ keys,
                                                    float* __restrict__ score) {
    __shared__ __align__(16) float As[64 * 36];
    __shared__ __align__(16) float Bs[128 * 36];
    const int tid = threadIdx.x;
    const int n0 = blockIdx.y * 64;
    const int m0 = blockIdx.x * 128;
    const int wave = tid >> 5, lane = tid & 31;
    const int lo = lane & 15, hi = lane >> 4;
    const int wn = wave & 1, wm = wave >> 1;

    v8f c[2][2] = {};
    for (int kk = 0; kk < Dd; kk += 32) {
        #pragma unroll
        for (int s = 0; s < 2; ++s) {               // A: 64x32 = 512 float4 slots
            int slot = tid + s * 256;
            int r = slot >> 3, c4 = (slot & 7) << 2;
            float4 v = *reinterpret_cast<const float4*>(&q[(size_t)(n0 + r) * Dd + kk + c4]);
            *reinterpret_cast<float4*>(&As[r * 36 + c4]) = v;
        }
        #pragma unroll
        for (int s = 0; s < 4; ++s) {               // B: 128x32 = 1024 float4 slots
            int slot = tid + s * 256;
            int r = slot >> 3, c4 = (slot & 7) << 2;
            float4 v = *reinterpret_cast<const float4*>(&keys[(size_t)(m0 + r) * Dd + kk + c4]);
            *reinterpret_cast<float4*>(&Bs[r * 36 + c4]) = v;
        }
        __syncthreads();
        #pragma unroll
        for (int ks = 0; ks < 32; ks += 4) {
            v2f a[2], bb[2];
            #pragma unroll
            for (int i = 0; i < 2; ++i) {
                int r = (wn * 32 + i * 16 + lo) * 36 + ks + 2 * hi;
                a[i].x = As[r];  a[i].y = As[r + 1];
            }
            #pragma unroll
            for (int j = 0; j < 2; ++j) {
                int r = (wm * 32 + j * 16 + lo) * 36 + ks + 2 * hi;
                bb[j].x = Bs[r]; bb[j].y = Bs[r + 1];
            }
            #pragma unroll
            for (int i = 0; i < 2; ++i)
                #pragma unroll
                for (int j = 0; j < 2; ++j)
                    c[i][j] = __builtin_amdgcn_wmma_f32_16x16x4_f32(
                        false, a[i], false, bb[j], (short)0, c[i][j], false, false);
        }
        __syncthreads();
    }
    #pragma unroll
    for (int i = 0; i < 2; ++i)
        #pragma unroll
        for (int j = 0; j < 2; ++j)
            #pragma unroll
            for (int v = 0; v < 8; ++v) {
                int n = n0 + wn * 32 + i * 16 + v + 8 * hi;
                int m = m0 + wm * 32 + j * 16 + lo;
                score[(size_t)n * Mm + m] = c[i][j][v];
            }
}

// ---------------------------------------------------------------- K3: colmax over n (raw score)
__global__ __launch_bounds__(256) void k_colmax(const float* __restrict__ score,
                                                unsigned* __restrict__ colmax_u) {
    const int m = blockIdx.x * 256 + threadIdx.x;   // gridDim.x = 4
    const int r0 = blockIdx.y * 256;                // gridDim.y = 64
    float mx = -__builtin_inff();
    for (int r = 0; r < 256; ++r)
        mx = fmaxf(mx, score[(size_t)(r0 + r) * Mm + m]);
    atomicMax(&colmax_u[m], enc_f(mx));
}

// ---------------------------------------------------------------- K4: row softmax (in-place) + top2 + wsel
__global__ __launch_bounds__(256) void k_rowsoftmax(float* score,
                                                    const unsigned* __restrict__ colmax_u,
                                                    int* __restrict__ top1,
                                                    int* __restrict__ top2,
                                                    float* __restrict__ wsel) {
    __shared__ float sv1[256], sv2[256], ssum[256];
    __shared__ int   si1[256], si2[256];
    const int n = blockIdx.x, t = threadIdx.x;
    float s[4];
    float v1 = -__builtin_inff(), v2 = -__builtin_inff();
    int i1 = Mm, i2 = Mm;
    #pragma unroll
    for (int k = 0; k < 4; ++k) {
        int m = t + k * 256;
        float val = score[(size_t)n * Mm + m];
        s[k] = val;
        if (val > v1 || (val == v1 && m < i1)) { v2 = v1; i2 = i1; v1 = val; i1 = m; }
        else if (val > v2 || (val == v2 && m < i2)) { v2 = val; i2 = m; }
    }
    sv1[t] = v1; si1[t] = i1; sv2[t] = v2; si2[t] = i2;
    __syncthreads();
    for (int off = 128; off > 0; off >>= 1) {
        if (t < off) {
            float a1 = sv1[t], a2 = sv2[t], b1 = sv1[t + off], b2 = sv2[t + off];
            int ai1 = si1[t], ai2 = si2[t], bi1 = si1[t + off], bi2 = si2[t + off];
            float n1, n2; int ni1, ni2;
            bool aw = (a1 > b1) || (a1 == b1 && ai1 < bi1);
            if (aw) {
                n1 = a1; ni1 = ai1;
                bool cw = (a2 > b1) || (a2 == b1 && ai2 < bi1);
                if (cw) { n2 = a2; ni2 = ai2; } else { n2 = b1; ni2 = bi1; }
            } else {
                n1 = b1; ni1 = bi1;
                bool cw = (b2 > a1) || (b2 == a1 && bi2 < ai1);
                if (cw) { n2 = b2; ni2 = bi2; } else { n2 = a1; ni2 = ai1; }
            }
            sv1[t] = n1; si1[t] = ni1; sv2[t] = n2; si2[t] = ni2;
        }
        __syncthreads();
    }
    const float rmax = sv1[0];
    const int t1 = si1[0], t2 = si2[0];
    float e[4], acc = 0.f;
    #pragma unroll
    for (int k = 0; k < 4; ++k) { e[k] = expf(s[k] - rmax); acc += e[k]; }
    ssum[t] = acc;
    __syncthreads();
    for (int off = 128; off > 0; off >>= 1) {
        if (t < off) ssum[t] += ssum[t + off];
        __syncthreads();
    }
    const float inv = 1.0f / ssum[0];
    #pragma unroll
    for (int k = 0; k < 4; ++k)
        score[(size_t)n * Mm + t + k * 256] = e[k] * inv;
    if (t == 0) {
        top1[n] = t1; top2[n] = t2;
        // wsel = score_query[n,t1]/colmax(score_query[:,t1]) = exp(score[n,t1]-colmax_raw[t1])
        wsel[n] = expf(rmax - dec_f(colmax_u[t1]));
    }
}

// ---------------------------------------------------------------- K5: per-row losses
__global__ __launch_bounds__(128) void k_loss(const float* __restrict__ q,
                                              const float* __restrict__ keys,
                                              const int* __restrict__ top1,
                                              const int* __restrict__ top2,
                                              float* __restrict__ tmp) {
    __shared__ float s0[128], s1[128], s2[128];
    const int n = blockIdx.x, t = threadIdx.x;
    const float* qr = q + (size_t)n * Dd;
    const float* pr = keys + (size_t)top1[n] * Dd;
    const float* gr = keys + (size_t)top2[n] * Dd;
    float a0 = 0.f, a1 = 0.f, a2 = 0.f;
    #pragma unroll
    for (int k = 0; k < 4; ++k) {
        int d = t + k * 128;
        float qv = qr[d], pv = pr[d], gv = gr[d];
        float dm = qv - pv;
        a0 += dm * dm;
        float da = dm + 1e-6f;      a1 += da * da;
        float dn = qv - gv + 1e-6f; a2 += dn * dn;
    }
    s0[t] = a0; s1[t] = a1; s2[t] = a2;
    __syncthreads();
    for (int off = 64; off > 0; off >>= 1) {
        if (t < off) { s0[t] += s0[t + off]; s1[t] += s1[t + off]; s2[t] += s2[t + off]; }
        __syncthreads();
    }
    if (t == 0) {
        tmp[n * 2 + 0] = s0[0];
        float trip = sqrtf(s1[0]) - sqrtf(s2[0]) + 1.0f;
        tmp[n * 2 + 1] = trip > 0.f ? trip : 0.f;
    }
}

// ---------------------------------------------------------------- K6: per-batch loss means (deterministic tree)
__global__ __launch_bounds__(256) void k_loss_reduce(const float* __restrict__ tmp,
                                                     float* __restrict__ out4) {
    __shared__ float s0[256], s1[256];
    const int b = blockIdx.x, t = threadIdx.x;
    float a0 = 0.f, a1 = 0.f;
    #pragma unroll
    for (int k = 0; k < 4; ++k) {
        int r = b * HWn + t + k * 256;
        a0 += tmp[r * 2 + 0];
        a1 += tmp[r * 2 + 1];
    }
    s0[t] = a0; s1[t] = a1;
    __syncthreads();
    for (int off = 128; off > 0; off >>= 1) {
        if (t < off) { s0[t] += s0[t + off]; s1[t] += s1[t + off]; }
        __syncthreads();
    }
    if (t == 0) {
        out4[b * 2 + 0] = s0[0] * (1.0f / HWn);
        out4[b * 2 + 1] = s1[0] * (1.0f / HWn);
    }
}

// ---------------------------------------------------------------- K7: concat_mem = P @ keys (fp32 WMMA), store transposed into out1 hi-channels
// block tile 64n x 128d, BK=32 over M=1024
__global__ __launch_bounds__(256) void k_gemm_concat(const float* __restrict__ P,
                                                     const float* __restrict__ keys,
                                                     float* __restrict__ out1) {
    __shared__ __align__(16) float As[64 * 36];
    __shared__ __align__(16) float Bs[32 * 132];
    __shared__ __align__(16) float Ct[128 * 65];
    const int tid = threadIdx.x;
    const int n0 = blockIdx.y * 64;
    const int d0 = blockIdx.x * 128;
    const int wave = tid >> 5, lane = tid & 31;
    const int lo = lane & 15, hi = lane >> 4;
    const int wn = wave & 1, wm = wave >> 1;

    v8f c[2][2] = {};
    for (int kk = 0; kk < Mm; kk += 32) {
        #pragma unroll
        for (int s = 0; s < 2; ++s) {               // A: P 64x32
            int slot = tid + s * 256;
            int r = slot >> 3, c4 = (slot & 7) << 2;
            float4 v = *reinterpret_cast<const float4*>(&P[(size_t)(n0 + r) * Mm + kk + c4]);
            *reinterpret_cast<float4*>(&As[r * 36 + c4]) = v;
        }
        #pragma unroll
        for (int s = 0; s < 4; ++s) {               // B: keys 32(k) x 128(d)
            int slot = tid + s * 256;
            int r = slot >> 5, c4 = (slot & 31) << 2;
            float4 v = *reinterpret_cast<const float4*>(&keys[(size_t)(kk + r) * Dd + d0 + c4]);
            *reinterpret_cast<float4*>(&Bs[r * 132 + c4]) = v;
        }
        __syncthreads();
        #pragma unroll
        for (int ks = 0; ks < 32; ks += 4) {
            v2f a[2], bb[2];
            #pragma unroll
            for (int i = 0; i < 2; ++i) {
                int r = (wn * 32 + i * 16 + lo) * 36 + ks + 2 * hi;
                a[i].x = As[r];  a[i].y = As[r + 1];
            }
            #pragma unroll
            for (int j = 0; j < 2; ++j) {
                int col = wm * 32 + j * 16 + lo;
                bb[j].x = Bs[(ks + 2 * hi) * 132 + col];
                bb[j].y = Bs[(ks + 2 * hi + 1) * 132 + col];
            }
            #pragma unroll
            for (int i = 0; i < 2; ++i)
                #pragma unroll
                for (int j = 0; j < 2; ++j)
                    c[i][j] = __builtin_amdgcn_wmma_f32_16x16x4_f32(
                        false, a[i], false, bb[j], (short)0, c[i][j], false, false);
        }
        __syncthreads();
    }
    // transpose through LDS so the channel-major stores are coalesced along hw
    #pragma unroll
    for (int i = 0; i < 2; ++i)
        #pragma unroll
        for (int j = 0; j < 2; ++j)
            #pragma unroll
            for (int v = 0; v < 8; ++v) {
                int dl = wm * 32 + j * 16 + lo;
                int nl = wn * 32 + i * 16 + v + 8 * hi;
                Ct[dl * 65 + nl] = c[i][j][v];
            }
    __syncthreads();
    const int bidx = n0 >> 10;
    const int hw0 = n0 & (HWn - 1);
    #pragma unroll
    for (int it = 0; it < 32; ++it) {
        int idx = tid + it * 256;
        int dl = idx >> 6, nl = idx & 63;
        out1[(size_t)bidx * (2 * Dd * HWn) + (size_t)(Dd + d0 + dl) * HWn + hw0 + nl] =
            Ct[dl * 65 + nl];
    }
}

// ---------------------------------------------------------------- K8: q -> out1 low channels (32x32 LDS transpose)
__global__ __launch_bounds__(256) void k_qwrite(const float* __restrict__ q,
                                                float* __restrict__ out1) {
    __shared__ float tl[32 * 33];
    const int t = threadIdx.x;
    const int n0 = blockIdx.y * 32, d0 = blockIdx.x * 32;
    const int cc = t & 31, r8 = t >> 5;
    #pragma unroll
    for (int i = 0; i < 4; ++i) {
        int r = r8 + i * 8;
        tl[r * 33 + cc] = q[(size_t)(n0 + r) * Dd + d0 + cc];
    }
    __syncthreads();
    const int b = n0 >> 10, hw0 = n0 & (HWn - 1);
    #pragma unroll
    for (int i = 0; i < 4; ++i) {
        int r = r8 + i * 8;   // d-row
        out1[(size_t)b * (2 * Dd * HWn) + (size_t)(d0 + r) * HWn + hw0 + cc] = tl[cc * 33 + r];
    }
}

// ---------------------------------------------------------------- K9: deterministic segment_sum + broadcast to all batches
// one block per memory slot m; fixed ascending-n scan order (bitwise deterministic)
__global__ __launch_bounds__(256) void k_segsum(const float* __restrict__ q,
                                                const int* __restrict__ top1,
                                                const float* __restrict__ wsel,
                                                float* __restrict__ out2) {
    __shared__ __align__(16) int tl[Nn];
    const int m = blockIdx.x, t = threadIdx.x;
    #pragma unroll
    for (int i = 0; i < 16; ++i) {
        int idx = t + i * 256;
        reinterpret_cast<int4*>(tl)[idx] = reinterpret_cast<const int4*>(top1)[idx];
    }
    __syncthreads();
    float a0 = 0.f, a1 = 0.f;
    const int d0 = t, d1 = t + 256;
    for (int n = 0; n < Nn; ++n) {
        if (tl[n] == m) {
            float w = wsel[n];
            const float* qr = q + (size_t)n * Dd;
            a0 += w * qr[d0];
            a1 += w * qr[d1];
        }
    }
    #pragma unroll
    for (int b = 0; b < Bn; ++b) {
        size_t base = (size_t)b * (Mm * Dd) + (size_t)m * Dd;
        out2[base + d0] = a0;
        out2[base + d1] = a1;
    }
}

// ================================================================ launch
extern "C" void kernel_launch(void* const* d_in, const int* in_sizes, int n_in,
                              void* d_out, int out_size, void* d_ws, size_t ws_size,
                              hipStream_t stream) {
    const float* query = (const float*)d_in[0];     // (B, D, H, W)
    const float* keys  = (const float*)d_in[1];     // (B, M, D); only batch 0 used
    float* out  = (float*)d_out;
    float* out1 = out;                               // updated_query  (16,1024,32,32)
    float* out2 = out + 16777216;                    // updated_memory (16,1024,512)
    float* out3 = out + 25165824;                    // score_memory_b (16,32,32,1024) == P (n x M)
    float* out4 = out + 41943040;                    // loss_list (16,2)

    char* ws = (char*)d_ws;
    float*    q_ws     = (float*)   (ws);                 // n x D = 32 MiB
    unsigned* colmax_u = (unsigned*)(ws + 33554432);      // M
    int*      top1     = (int*)     (ws + 33558528);      // n
    int*      top2     = (int*)     (ws + 33624064);      // n
    float*    wsel     = (float*)   (ws + 33689600);      // n
    float*    tmp_loss = (float*)   (ws + 33755136);      // n x 2

    k_init<<<4, 256, 0, stream>>>(colmax_u);
    k_normalize<<<Nn / 32, 256, 0, stream>>>(query, q_ws);
    k_gemm_score<<<dim3(Mm / 128, Nn / 64), 256, 0, stream>>>(q_ws, keys, out3);
    k_colmax<<<dim3(4, 64), 256, 0, stream>>>(out3, colmax_u);
    k_rowsoftmax<<<Nn, 256, 0, stream>>>(out3, colmax_u, top1, top2, wsel);
    k_loss<<<Nn, 128, 0, stream>>>(q_ws, keys, top1, top2, tmp_loss);
    k_loss_reduce<<<Bn, 256, 0, stream>>>(tmp_loss, out4);
    k_gemm_concat<<<dim3(Dd / 128, Nn / 64), 256, 0, stream>>>(out3, keys, out1);
    k_qwrite<<<dim3(Dd / 32, Nn / 32), 256, 0, stream>>>(q_ws, out1);
    k_segsum<<<Mm, 256, 0, stream>>>(q_ws, top1, wsel, out2);
}